// PC_CNN_48584670052802
// MI455X (gfx1250) — compile-verified
//
#include <hip/hip_runtime.h>
#include <math.h>

typedef __attribute__((ext_vector_type(2))) float v2f;
typedef __attribute__((ext_vector_type(8))) float v8f;

#define LR_R 0.02f

// ---------------------------------------------------------------------------
// Weight repack: OIHW (Cout,Cin,3,3) -> [tap][Cin][Cout]  (GEMM-B friendly)
// ---------------------------------------------------------------------------
__global__ void pack_oihw(const float* __restrict__ w, float* __restrict__ wp,
                          int Cout, int Cin) {
  int idx = blockIdx.x * blockDim.x + threadIdx.x;
  int total = Cout * Cin * 9;
  if (idx >= total) return;
  int co = idx % Cout;
  int ci = (idx / Cout) % Cin;
  int t  = idx / (Cout * Cin);
  wp[idx] = w[(co * Cin + ci) * 9 + t];
}

// ConvTranspose weight (CinT,CoutT,3,3) -> equivalent conv weight packed
// [tap][CinT][CoutT], with spatial flip: wc[o][i][ky][kx] = P[i][o][2-ky][2-kx]
__global__ void pack_transposed(const float* __restrict__ p, float* __restrict__ wp,
                                int CinT, int CoutT) {
  int idx = blockIdx.x * blockDim.x + threadIdx.x;
  int total = CinT * CoutT * 9;
  if (idx >= total) return;
  int o = idx % CoutT;
  int i = (idx / CoutT) % CinT;
  int t = idx / (CoutT * CinT);
  wp[idx] = p[(i * CoutT + o) * 9 + (8 - t)];
}

// ---------------------------------------------------------------------------
// Layer-1 conv: Cin=3 (K=27, not WMMA-friendly; ~0.9 GFLOP, direct VALU)
// x [B,3,32,32], W1 [64,3,3,3] -> y1 [B,64,32,32]
// ---------------------------------------------------------------------------
__global__ void conv1_direct(const float* __restrict__ x, const float* __restrict__ W1,
                             float* __restrict__ y1, int B) {
  int idx = blockIdx.x * blockDim.x + threadIdx.x;
  int total = B * 64 * 32 * 32;
  if (idx >= total) return;
  int xw = idx & 31;
  int yh = (idx >> 5) & 31;
  int co = (idx >> 10) & 63;
  int n  = idx >> 16;
  float acc = 0.f;
  #pragma unroll
  for (int ci = 0; ci < 3; ++ci) {
    const float* px = x + ((n * 3 + ci) * 32) * 32;
    const float* pw = W1 + ((co * 3 + ci) * 3) * 3;
    #pragma unroll
    for (int ky = 0; ky < 3; ++ky) {
      int iy = yh + ky - 1;
      if (iy < 0 || iy > 31) continue;
      #pragma unroll
      for (int kx = 0; kx < 3; ++kx) {
        int ix = xw + kx - 1;
        if (ix < 0 || ix > 31) continue;
        acc += px[iy * 32 + ix] * pw[ky * 3 + kx];
      }
    }
  }
  y1[idx] = acc;
}

// ---------------------------------------------------------------------------
// BatchNorm training-mode statistics: one block per channel.
// y [B,C,H,W] -> mean[c], inv[c] = rsqrt(var+eps), biased variance.
// ---------------------------------------------------------------------------
__global__ void bn_stats(const float* __restrict__ y, float* __restrict__ mean,
                         float* __restrict__ inv, int C, int HW, int B) {
  int c = blockIdx.x;
  int tid = threadIdx.x;
  int total = B * HW;
  float s = 0.f, s2 = 0.f;
  for (int i = tid; i < total; i += blockDim.x) {
    int n = i / HW;
    int sp = i - n * HW;
    float v = y[(n * C + c) * HW + sp];
    s += v;
    s2 += v * v;
  }
  __shared__ float sh[256], sh2[256];
  sh[tid] = s; sh2[tid] = s2;
  __syncthreads();
  for (int off = 128; off > 0; off >>= 1) {
    if (tid < off) { sh[tid] += sh[tid + off]; sh2[tid] += sh2[tid + off]; }
    __syncthreads();
  }
  if (tid == 0) {
    float invN = 1.0f / (float)total;
    float m = sh[0] * invN;
    mean[c] = m;
    inv[c] = rsqrtf(sh2[0] * invN - m * m + 1e-5f);
  }
}

// ---------------------------------------------------------------------------
// BN-apply + ReLU + 2x2 maxpool.  y [B,C,H,H] -> r [B,C,H/2,H/2]
// ---------------------------------------------------------------------------
__global__ void bn_relu_pool(const float* __restrict__ y, const float* __restrict__ g,
                             const float* __restrict__ b, const float* __restrict__ mean,
                             const float* __restrict__ inv, float* __restrict__ r,
                             int B, int C, int H) {
  int Ho = H >> 1;
  int idx = blockIdx.x * blockDim.x + threadIdx.x;
  int total = B * C * Ho * Ho;
  if (idx >= total) return;
  int xx = idx % Ho;
  int yy = (idx / Ho) % Ho;
  int c  = (idx / (Ho * Ho)) % C;
  int n  = idx / (Ho * Ho * C);
  const float* p = y + ((n * C + c) * H + 2 * yy) * H + 2 * xx;
  float sc = g[c] * inv[c];
  float sb = b[c] - mean[c] * sc;
  float v0 = fmaxf(p[0]     * sc + sb, 0.f);
  float v1 = fmaxf(p[1]     * sc + sb, 0.f);
  float v2 = fmaxf(p[H]     * sc + sb, 0.f);
  float v3 = fmaxf(p[H + 1] * sc + sb, 0.f);
  r[idx] = fmaxf(fmaxf(v0, v1), fmaxf(v2, v3));
}

// ---------------------------------------------------------------------------
// Implicit-GEMM 3x3 conv (pad=1, stride=1) on V_WMMA_F32_16X16X4_F32.
// One wave per block computes (16*MT) spatial positions x (16*NT) couts.
// Per K-step(4): 2*MT A-loads + 2*NT B-loads feed MT*NT WMMAs (2 loads/WMMA
// at MT=NT=2).  All loads are through integer offsets off a single
// kernel-arg base pointer (keeps them in the GLOBAL address space; no
// flat_load, no 64-bit pointer bump chains).  Halo handling is branch-free
// (clamped offset + v_cndmask zeroing).
// A layout (16x4): lanes 0-15 -> M=lane, K={k,k+1}; lanes 16-31 -> K={k+2,k+3}
// B layout (4x16): same K halving, lanes index N.
// D layout: VGPR r -> M=r (lanes 0-15) / M=r+8 (lanes 16-31), N=lane&15.
// MODE 0: store conv output.
// MODE 1: t=tanh(acc); up2 to 2H x 2W; e1 = r1 - t; r1 = relu(r1 - LR*e1).
// MODE 2: t=tanh(acc); up2; e2 = r2 - t.
// ---------------------------------------------------------------------------
template<int MODE, int MT, int NT, int Cin, int H, int W, int Cout>
__global__ void conv3x3_wmma(const float* __restrict__ in, const float* __restrict__ wp,
                             float* __restrict__ out, const float* __restrict__ raux,
                             float* __restrict__ eaux) {
  constexpr int HW = H * W;
  constexpr int PT = HW / (16 * MT);      // M-tiles per image
  const int n = blockIdx.x / PT;
  const int p0 = (blockIdx.x % PT) * (16 * MT);
  const int cout0 = blockIdx.y * (16 * NT);
  const int lane = threadIdx.x;
  const int li = lane & 15;
  const int khalf = (lane >> 4) << 1;     // 0 or 2
  const int mhalf = (lane >> 4) << 3;     // 0 or 8

  const int inBase = n * (Cin * HW);      // element offset of this image

  v8f acc[MT][NT];
  #pragma unroll
  for (int i = 0; i < MT; ++i)
    #pragma unroll
    for (int j = 0; j < NT; ++j)
      acc[i][j] = (v8f){0.f, 0.f, 0.f, 0.f, 0.f, 0.f, 0.f, 0.f};

  int oy[MT], ox[MT];
  #pragma unroll
  for (int i = 0; i < MT; ++i) {
    int m = p0 + i * 16 + li;
    oy[i] = m / W;
    ox[i] = m % W;
  }

  #pragma unroll
  for (int t = 0; t < 9; ++t) {
    const int dy = t / 3 - 1;
    const int dx = t % 3 - 1;

    int offA[MT];
    bool vld[MT];
    #pragma unroll
    for (int i = 0; i < MT; ++i) {
      int iy = oy[i] + dy;
      int ix = ox[i] + dx;
      vld[i] = ((unsigned)iy < (unsigned)H) & ((unsigned)ix < (unsigned)W);
      int iyc = min(max(iy, 0), H - 1);   // clamped: load always in-bounds
      int ixc = min(max(ix, 0), W - 1);
      offA[i] = inBase + khalf * HW + iyc * W + ixc;
    }
    int offB = (t * Cin + khalf) * Cout + cout0 + li;

    for (int k = 0; k < Cin; k += 4) {
      v2f b[NT];
      #pragma unroll
      for (int j = 0; j < NT; ++j) {
        b[j].x = wp[offB + j * 16];
        b[j].y = wp[offB + j * 16 + Cout];
      }
      offB += 4 * Cout;

      v2f a[MT];
      #pragma unroll
      for (int i = 0; i < MT; ++i) {
        float a0 = in[offA[i]];           // unconditional global load
        float a1 = in[offA[i] + HW];
        offA[i] += 4 * HW;
        a[i].x = vld[i] ? a0 : 0.f;       // v_cndmask, no exec divergence
        a[i].y = vld[i] ? a1 : 0.f;
      }

      #pragma unroll
      for (int i = 0; i < MT; ++i)
        #pragma unroll
        for (int j = 0; j < NT; ++j)
          acc[i][j] = __builtin_amdgcn_wmma_f32_16x16x4_f32(
              false, a[i], false, b[j], (short)0, acc[i][j], false, false);
    }
  }

  #pragma unroll
  for (int i = 0; i < MT; ++i) {
    #pragma unroll
    for (int j = 0; j < NT; ++j) {
      const int cN = cout0 + j * 16 + li;
      #pragma unroll
      for (int r = 0; r < 8; ++r) {
        int mm = p0 + i * 16 + r + mhalf;
        int yy = mm / W;
        int xx = mm % W;
        float val = acc[i][j][r];
        if (MODE == 0) {
          out[((n * Cout + cN) * H + yy) * W + xx] = val;
        } else {
          float tv = tanhf(val);
          constexpr int H2 = H * 2;
          int base = ((n * Cout + cN) * H2 + 2 * yy) * H2 + 2 * xx;
          const int offs[4] = {0, 1, H2, H2 + 1};
          #pragma unroll
          for (int s = 0; s < 4; ++s) {
            int o = base + offs[s];
            float rv = raux[o];
            float ev = rv - tv;
            eaux[o] = ev;
            if (MODE == 1) {
              // dr1 = -e1; r1 = relu(r1 + LR*dr1)
              ((float*)raux)[o] = fmaxf(rv - LR_R * ev, 0.f);
            }
          }
        }
      }
    }
  }
}

// ---------------------------------------------------------------------------
// r2 = relu(r2 + LR*(-e2 + avgpool2(t2)));  t2 [B,128,16,16], r2/e2 [B,128,8,8]
// ---------------------------------------------------------------------------
__global__ void update_r2(const float* __restrict__ t2, const float* __restrict__ e2,
                          float* __restrict__ r2, int B) {
  int idx = blockIdx.x * blockDim.x + threadIdx.x;
  int total = B * 128 * 64;
  if (idx >= total) return;
  int x = idx & 7;
  int y = (idx >> 3) & 7;
  int c = (idx >> 6) & 127;
  int n = idx >> 13;
  const float* p = t2 + ((n * 128 + c) * 16 + 2 * y) * 16 + 2 * x;
  float avg = 0.25f * (p[0] + p[1] + p[16] + p[17]);
  r2[idx] = fmaxf(r2[idx] + LR_R * (avg - e2[idx]), 0.f);
}

// ---------------------------------------------------------------------------
// r3 = relu(r3 + LR*avgpool2(t3));  t3 [B,256,8,8], r3 [B,256,4,4]
// ---------------------------------------------------------------------------
__global__ void update_r3(const float* __restrict__ t3, float* __restrict__ r3, int B) {
  int idx = blockIdx.x * blockDim.x + threadIdx.x;
  int total = B * 256 * 16;
  if (idx >= total) return;
  int x = idx & 3;
  int y = (idx >> 2) & 3;
  int c = (idx >> 4) & 255;
  int n = idx >> 12;
  const float* p = t3 + ((n * 256 + c) * 8 + 2 * y) * 8 + 2 * x;
  float avg = 0.25f * (p[0] + p[1] + p[8] + p[9]);
  r3[idx] = fmaxf(r3[idx] + LR_R * avg, 0.f);
}

// ---------------------------------------------------------------------------
extern "C" void kernel_launch(void* const* d_in, const int* in_sizes, int n_in,
                              void* d_out, int out_size, void* d_ws, size_t ws_size,
                              hipStream_t stream) {
  const float* x  = (const float*)d_in[0];
  const float* W1 = (const float*)d_in[1];
  const float* W2 = (const float*)d_in[2];
  const float* W3 = (const float*)d_in[3];
  const float* P2 = (const float*)d_in[4];
  const float* P3 = (const float*)d_in[5];
  const float* g1 = (const float*)d_in[6];
  const float* b1 = (const float*)d_in[7];
  const float* g2 = (const float*)d_in[8];
  const float* b2 = (const float*)d_in[9];
  const float* g3 = (const float*)d_in[10];
  const float* b3 = (const float*)d_in[11];

  const int B = 256;

  // outputs (r1|r2|r3 concatenated, computed in place)
  float* r1 = (float*)d_out;                  // [256,64,16,16]  = 4194304
  float* r2 = r1 + 4194304;                   // [256,128,8,8]   = 2097152
  float* r3 = r2 + 2097152;                   // [256,256,4,4]   = 1048576

  // workspace layout (floats)
  float* ws   = (float*)d_ws;
  float* y1   = ws;                           // [256,64,32,32]  = 16777216
  float* y2   = y1 + 16777216;                // [256,128,16,16] = 8388608
  float* y3   = y2 + 8388608;                 // [256,256,8,8]   = 4194304
  float* w2p  = y3 + 4194304;                 // 9*64*128  = 73728
  float* w3p  = w2p + 73728;                  // 9*128*256 = 294912
  float* w2tp = w3p + 294912;                 // 9*128*64  = 73728
  float* w3tp = w2tp + 73728;                 // 9*256*128 = 294912
  float* mean = w3tp + 294912;                // 256
  float* inv  = mean + 256;                   // 256
  // aliases (feed-forward scratch no longer needed during iterations)
  float* e1 = y1;                             // [256,64,16,16]
  float* e2 = y1 + 4194304;                   // [256,128,8,8]
  float* t2 = y2;                             // [256,128,16,16]
  float* t3 = y3;                             // [256,256,8,8]

  // ---- weight repack ----
  pack_oihw<<<(9 * 64 * 128 + 255) / 256, 256, 0, stream>>>(W2, w2p, 128, 64);
  pack_oihw<<<(9 * 128 * 256 + 255) / 256, 256, 0, stream>>>(W3, w3p, 256, 128);
  pack_transposed<<<(9 * 128 * 64 + 255) / 256, 256, 0, stream>>>(P2, w2tp, 128, 64);
  pack_transposed<<<(9 * 256 * 128 + 255) / 256, 256, 0, stream>>>(P3, w3tp, 256, 128);

  // ---- feed-forward init ----
  conv1_direct<<<(B * 65536 + 255) / 256, 256, 0, stream>>>(x, W1, y1, B);
  bn_stats<<<64, 256, 0, stream>>>(y1, mean, inv, 64, 1024, B);
  bn_relu_pool<<<(B * 64 * 256 + 255) / 256, 256, 0, stream>>>(y1, g1, b1, mean, inv, r1, B, 64, 32);

  // conv2: r1 [B,64,16,16] -> y2 [B,128,16,16]   (HW=256: PT=8, Couttiles=4)
  conv3x3_wmma<0, 2, 2, 64, 16, 16, 128>
      <<<dim3(B * 8, 4), 32, 0, stream>>>(r1, w2p, y2, nullptr, nullptr);
  bn_stats<<<128, 256, 0, stream>>>(y2, mean, inv, 128, 256, B);
  bn_relu_pool<<<(B * 128 * 64 + 255) / 256, 256, 0, stream>>>(y2, g2, b2, mean, inv, r2, B, 128, 16);

  // conv3: r2 [B,128,8,8] -> y3 [B,256,8,8]      (HW=64: PT=2, Couttiles=8)
  conv3x3_wmma<0, 2, 2, 128, 8, 8, 256>
      <<<dim3(B * 2, 8), 32, 0, stream>>>(r2, w3p, y3, nullptr, nullptr);
  bn_stats<<<256, 256, 0, stream>>>(y3, mean, inv, 256, 64, B);
  bn_relu_pool<<<(B * 256 * 16 + 255) / 256, 256, 0, stream>>>(y3, g3, b3, mean, inv, r3, B, 256, 8);

  // ---- predictive-coding iterations ----
  for (int it = 0; it < 10; ++it) {
    // pred1 = tanh(up2(convT(r2,P2))); e1 = r1 - pred1; r1 = relu(r1 - LR*e1)
    conv3x3_wmma<1, 2, 2, 128, 8, 8, 64>
        <<<dim3(B * 2, 2), 32, 0, stream>>>(r2, w2tp, nullptr, r1, e1);
    // pred2 = tanh(up2(convT(r3,P3))); e2 = r2 - pred2   (HW=16 -> MT=1)
    conv3x3_wmma<2, 1, 2, 256, 4, 4, 128>
        <<<dim3(B * 1, 4), 32, 0, stream>>>(r3, w3tp, nullptr, r2, e2);
    // t2 = conv(e1, W2) at 16x16
    conv3x3_wmma<0, 2, 2, 64, 16, 16, 128>
        <<<dim3(B * 8, 4), 32, 0, stream>>>(e1, w2p, t2, nullptr, nullptr);
    // r2 = relu(r2 + LR*(-e2 + avgpool2(t2)))
    update_r2<<<(B * 128 * 64 + 255) / 256, 256, 0, stream>>>(t2, e2, r2, B);
    // t3 = conv(e2, W3) at 8x8
    conv3x3_wmma<0, 2, 2, 128, 8, 8, 256>
        <<<dim3(B * 2, 8), 32, 0, stream>>>(e2, w3p, t3, nullptr, nullptr);
    // r3 = relu(r3 + LR*avgpool2(t3))
    update_r3<<<(B * 256 * 16 + 255) / 256, 256, 0, stream>>>(t3, r3, B);
  }
}